// RWKV_RNN_76630806495664
// MI455X (gfx1250) — compile-verified
//
#include <hip/hip_runtime.h>
#include <hip/hip_bf16.h>
#include <cstddef>

// ---------------------------------------------------------------------------
// RWKV-7 single-token forward for MI455X (gfx1250, wave32).
// Bandwidth-bound (~810MB fp32 weights/token -> ~35us floor @23.3TB/s).
// Big GEMVs: coalesced float4 streaming + global_prefetch.
// State update S = S*w + S@ab + v(x)k, out = S@r: per-head 64x64x64 fp32
// matmul done with V_WMMA_F32_16X16X4_F32 (one wave per head).
// ---------------------------------------------------------------------------

constexpr int kL  = 12;
constexpr int kC  = 1024;
constexpr int kH  = 16;
constexpr int kN  = 64;
constexpr int kV  = 50304;
constexpr int kLW = 64;
constexpr int kLA = 64;
constexpr int kLV = 32;
constexpr int kLG = 128;
constexpr int kFF = 4096;

typedef float v2f __attribute__((ext_vector_type(2)));
typedef float v8f __attribute__((ext_vector_type(8)));

__device__ __forceinline__ float sigf(float x) { return 1.0f / (1.0f + expf(-x)); }

// ---------------------------------------------------------------------------
// x[c] = emb[token*C + c]
// ---------------------------------------------------------------------------
__global__ __launch_bounds__(256) void embed_kernel(const float* __restrict__ emb,
                                                    const int* __restrict__ token,
                                                    float* __restrict__ x) {
  int c = blockIdx.x * 256 + threadIdx.x;
  if (c < kC) x[c] = emb[(size_t)token[0] * kC + c];
}

// ---------------------------------------------------------------------------
// Optional residual add, layernorm, then 6-way token-shift mix (time-mix pre).
// Writes na_out = ln(x).
// ---------------------------------------------------------------------------
__global__ __launch_bounds__(256) void ln_mix6_kernel(
    float* __restrict__ x, const float* __restrict__ res,
    const float* __restrict__ lnw, const float* __restrict__ lnb,
    const float* __restrict__ xprev,
    const float* __restrict__ m0, const float* __restrict__ m1,
    const float* __restrict__ m2, const float* __restrict__ m3,
    const float* __restrict__ m4, const float* __restrict__ m5,
    float* __restrict__ o0, float* __restrict__ o1, float* __restrict__ o2,
    float* __restrict__ o3, float* __restrict__ o4, float* __restrict__ o5,
    float* __restrict__ na_out) {
  __shared__ float rs[256], rq[256];
  __shared__ float s_mu, s_inv;
  int t = threadIdx.x;
  if (res) {
    for (int c = t; c < kC; c += 256) x[c] += res[c];
    __syncthreads();
  }
  float s1 = 0.f, s2 = 0.f;
  for (int c = t; c < kC; c += 256) { float xv = x[c]; s1 += xv; s2 += xv * xv; }
  rs[t] = s1; rq[t] = s2;
  __syncthreads();
  for (int o = 128; o > 0; o >>= 1) {
    if (t < o) { rs[t] += rs[t + o]; rq[t] += rq[t + o]; }
    __syncthreads();
  }
  if (t == 0) {
    float mu = rs[0] / (float)kC;
    s_mu = mu;
    s_inv = rsqrtf(rq[0] / (float)kC - mu * mu + 1e-5f);
  }
  __syncthreads();
  float mu = s_mu, inv = s_inv;
  for (int c = t; c < kC; c += 256) {
    float xx = (x[c] - mu) * inv * lnw[c] + lnb[c];
    na_out[c] = xx;
    float dx = xprev[c] - xx;
    o0[c] = xx + dx * m0[c];
    o1[c] = xx + dx * m1[c];
    o2[c] = xx + dx * m2[c];
    o3[c] = xx + dx * m3[c];
    o4[c] = xx + dx * m4[c];
    o5[c] = xx + dx * m5[c];
  }
}

// ---------------------------------------------------------------------------
// x += res; ln2; single token-shift mix (channel-mix pre). Writes nf_out.
// ---------------------------------------------------------------------------
__global__ __launch_bounds__(256) void ln_mix1_kernel(
    float* __restrict__ x, const float* __restrict__ res,
    const float* __restrict__ lnw, const float* __restrict__ lnb,
    const float* __restrict__ xprev, const float* __restrict__ mix,
    float* __restrict__ nf_out, float* __restrict__ out) {
  __shared__ float rs[256], rq[256];
  __shared__ float s_mu, s_inv;
  int t = threadIdx.x;
  for (int c = t; c < kC; c += 256) x[c] += res[c];
  __syncthreads();
  float s1 = 0.f, s2 = 0.f;
  for (int c = t; c < kC; c += 256) { float xv = x[c]; s1 += xv; s2 += xv * xv; }
  rs[t] = s1; rq[t] = s2;
  __syncthreads();
  for (int o = 128; o > 0; o >>= 1) {
    if (t < o) { rs[t] += rs[t + o]; rq[t] += rq[t + o]; }
    __syncthreads();
  }
  if (t == 0) {
    float mu = rs[0] / (float)kC;
    s_mu = mu;
    s_inv = rsqrtf(rq[0] / (float)kC - mu * mu + 1e-5f);
  }
  __syncthreads();
  float mu = s_mu, inv = s_inv;
  for (int c = t; c < kC; c += 256) {
    float xx = (x[c] - mu) * inv * lnw[c] + lnb[c];
    nf_out[c] = xx;
    out[c] = xx + (xprev[c] - xx) * mix[c];
  }
}

// ---------------------------------------------------------------------------
// x += res; final layernorm -> out
// ---------------------------------------------------------------------------
__global__ __launch_bounds__(256) void ln_final_kernel(
    float* __restrict__ x, const float* __restrict__ res,
    const float* __restrict__ lnw, const float* __restrict__ lnb,
    float* __restrict__ out) {
  __shared__ float rs[256], rq[256];
  __shared__ float s_mu, s_inv;
  int t = threadIdx.x;
  for (int c = t; c < kC; c += 256) x[c] += res[c];
  __syncthreads();
  float s1 = 0.f, s2 = 0.f;
  for (int c = t; c < kC; c += 256) { float xv = x[c]; s1 += xv; s2 += xv * xv; }
  rs[t] = s1; rq[t] = s2;
  __syncthreads();
  for (int o = 128; o > 0; o >>= 1) {
    if (t < o) { rs[t] += rs[t + o]; rq[t] += rq[t + o]; }
    __syncthreads();
  }
  if (t == 0) {
    float mu = rs[0] / (float)kC;
    s_mu = mu;
    s_inv = rsqrtf(rq[0] / (float)kC - mu * mu + 1e-5f);
  }
  __syncthreads();
  float mu = s_mu, inv = s_inv;
  for (int c = t; c < kC; c += 256)
    out[c] = (x[c] - mu) * inv * lnw[c] + lnb[c];
}

// ---------------------------------------------------------------------------
// y = act(W @ x [+ bias]).  One wave per output row, 8 rows per block.
// Coalesced float4 weight streaming + global prefetch (bandwidth-bound path).
// act: 0=none, 1=tanh, 2=sigmoid, 3=relu^2
// ---------------------------------------------------------------------------
__global__ __launch_bounds__(256) void gemv_kernel(
    const float* __restrict__ W, const float* __restrict__ x,
    float* __restrict__ y, const float* __restrict__ bias,
    int M, int K, int act) {
  int wave = threadIdx.x >> 5;
  int lane = threadIdx.x & 31;
  int row = blockIdx.x * 8 + wave;
  const float* Wr = W + (size_t)row * (size_t)K;
  float s = 0.f;
  for (int k0 = lane * 4; k0 < K; k0 += 128) {
    __builtin_prefetch(Wr + k0 + 1024, 0, 1);  // global_prefetch_b8 lookahead
    float4 wv = *(const float4*)(Wr + k0);
    float4 xv = *(const float4*)(x + k0);
    s += wv.x * xv.x + wv.y * xv.y + wv.z * xv.z + wv.w * xv.w;
  }
  __shared__ float red[256];
  red[threadIdx.x] = s;
  __syncthreads();
  for (int o = 16; o > 0; o >>= 1) {
    if (lane < o) red[threadIdx.x] += red[threadIdx.x + o];
    __syncthreads();
  }
  if (lane == 0) {
    float r = red[wave * 32];
    if (bias) r += bias[row];
    if (act == 1) r = tanhf(r);
    else if (act == 2) r = sigf(r);
    else if (act == 3) { float u = fmaxf(r, 0.f); r = u * u; }
    y[row] = r;
  }
}

// ---------------------------------------------------------------------------
// Elementwise pre-state fixups:
//   kk = normalize_per_head(k * k_k); k *= 1+(a-1)*k_a;
//   v_first handling; w = exp(-0.606531*sigmoid(w0_base + w))
// ---------------------------------------------------------------------------
__global__ __launch_bounds__(256) void prep_state_kernel(
    float* __restrict__ k, const float* __restrict__ k_k,
    const float* __restrict__ a, const float* __restrict__ k_a,
    float* __restrict__ v, const float* __restrict__ vgate,
    float* __restrict__ v_first, float* __restrict__ w,
    const float* __restrict__ w0_base, float* __restrict__ kk_out,
    int is_first) {
  __shared__ float ss[kH];
  int t = threadIdx.x;
  if (t < kH) ss[t] = 0.f;
  __syncthreads();
  for (int c = t; c < kC; c += 256) {
    float kv = k[c] * k_k[c];
    kk_out[c] = kv;
    atomicAdd(&ss[c >> 6], kv * kv);
  }
  __syncthreads();
  for (int c = t; c < kC; c += 256) {
    float inv = 1.f / fmaxf(sqrtf(ss[c >> 6]), 1e-12f);
    kk_out[c] *= inv;
    float av = a[c];
    k[c] = k[c] * (1.f + (av - 1.f) * k_a[c]);
    if (is_first) {
      v_first[c] = v[c];
    } else {
      float vv = v[c];
      v[c] = vv + (v_first[c] - vv) * vgate[c];
    }
    w[c] = expf(-0.606531f * sigf(w0_base[c] + w[c]));
  }
}

// ---------------------------------------------------------------------------
// Per-head state update with V_WMMA_F32_16X16X4_F32:
//   S_new = S * w[None,:] + S @ ab + v (x) k  (ab[j,c] = -kk[j]*(kk[c]*a[c]))
//   out   = groupnorm(S_new @ r) + (sum_j r*k*r_k)*v, then * g
// One wave (32 lanes) per head; S head tile (16KB) staged in LDS.
// ---------------------------------------------------------------------------
__global__ __launch_bounds__(32) void state_update_kernel(
    const float* __restrict__ S_in, float* __restrict__ S_out,
    const float* __restrict__ r, const float* __restrict__ k,
    const float* __restrict__ v, const float* __restrict__ kk,
    const float* __restrict__ a, const float* __restrict__ w,
    float* __restrict__ att, const float* __restrict__ gn_w,
    const float* __restrict__ gn_b, const float* __restrict__ r_k,
    const float* __restrict__ g) {
  int h = blockIdx.x;
  int lane = threadIdx.x;
  __shared__ __align__(16) float Sl[kN * kN];
  __shared__ float rl[kN], kl[kN], vl[kN], wl[kN], negkk[kN], kka[kN];
  __shared__ float outacc[kN];
  __shared__ float rA[32], rB[32], rD[32];

  const float* Sg = S_in + (size_t)h * kN * kN;
  for (int i = lane; i < kN * kN / 4; i += 32)
    ((float4*)Sl)[i] = ((const float4*)Sg)[i];
  for (int i = lane; i < kN; i += 32) {
    int c = h * kN + i;
    rl[i] = r[c]; kl[i] = k[c]; vl[i] = v[c]; wl[i] = w[c];
    float kv = kk[c];
    negkk[i] = -kv;
    kka[i] = kv * a[c];
    outacc[i] = 0.f;
  }
  __syncthreads();

  const int hi2 = lane >> 4;   // lane half (0/1)
  const int lan = lane & 15;

  for (int it = 0; it < 4; ++it) {
    for (int jt = 0; jt < 4; ++jt) {
      // C preload: S*w (column scale) + v (outer) k
      v8f acc;
#pragma unroll
      for (int t = 0; t < 8; ++t) {
        int i = it * 16 + t + hi2 * 8;
        int j = jt * 16 + lan;
        acc[t] = Sl[i * kN + j] * wl[j] + vl[i] * kl[j];
      }
      int bn = jt * 16 + lan;
      float kkan = kka[bn];
#pragma unroll
      for (int kb = 0; kb < 16; ++kb) {
        int k0 = kb * 4;
        int ac = k0 + hi2 * 2;
        // A tile (16x4 of S): lanes 0-15 -> K{0,1}, lanes 16-31 -> K{2,3}
        int ai = it * 16 + lan;
        v2f A;
        A[0] = Sl[ai * kN + ac];
        A[1] = Sl[ai * kN + ac + 1];
        // B tile (4x16 of ab), rank-1: -kk[j] * (kk*a)[c]
        v2f B;
        B[0] = negkk[ac] * kkan;
        B[1] = negkk[ac + 1] * kkan;
        acc = __builtin_amdgcn_wmma_f32_16x16x4_f32(
            false, A, false, B, (short)0, acc, false, false);
      }
      // Store S_new tile and accumulate out = S_new @ r
#pragma unroll
      for (int t = 0; t < 8; ++t) {
        int i = it * 16 + t + hi2 * 8;
        int j = jt * 16 + lan;
        S_out[(size_t)h * kN * kN + i * kN + j] = acc[t];
        atomicAdd(&outacc[i], acc[t] * rl[j]);
      }
    }
  }
  __syncthreads();

  // group-norm over the 64 head outputs + (r*k*r_k).sum * v term, then * g
  float s1 = 0.f, s2 = 0.f, dot = 0.f;
  for (int i = lane; i < kN; i += 32) {
    float ov = outacc[i];
    s1 += ov; s2 += ov * ov;
    dot += rl[i] * kl[i] * r_k[h * kN + i];
  }
  rA[lane] = s1; rB[lane] = s2; rD[lane] = dot;
  __syncthreads();
  if (lane == 0) {
    for (int t = 1; t < 32; ++t) { rA[0] += rA[t]; rB[0] += rB[t]; rD[0] += rD[t]; }
  }
  __syncthreads();
  float mu = rA[0] / (float)kN;
  float var = rB[0] / (float)kN - mu * mu;
  float inv = rsqrtf(var + 6.4e-4f);
  float dotv = rD[0];
  for (int i = lane; i < kN; i += 32) {
    int c = h * kN + i;
    float ov = (outacc[i] - mu) * inv * gn_w[c] + gn_b[c];
    ov += dotv * vl[i];
    att[c] = ov * g[c];
  }
}

// ---------------------------------------------------------------------------
// Host-side orchestration
// ---------------------------------------------------------------------------
extern "C" void kernel_launch(void* const* d_in, const int* in_sizes, int n_in,
                              void* d_out, int out_size, void* d_ws, size_t ws_size,
                              hipStream_t stream) {
  auto F = [&](int i) { return (const float*)d_in[i]; };
  const int* token = (const int*)d_in[0];
  const float* emb    = F(1);
  const float* head_w = F(2);
  const float* lnow   = F(3);
  const float* lnob   = F(4);
  const float* ln1w   = F(5);
  const float* ln1b   = F(6);
  const float* ln2w   = F(7);
  const float* ln2b   = F(8);
  const float* xrm    = F(9);
  const float* xwm    = F(10);
  const float* xkm    = F(11);
  const float* xvm    = F(12);
  const float* xam    = F(13);
  const float* xgm    = F(14);
  const float* w0     = F(15);
  const float* w1     = F(16);
  const float* w2     = F(17);
  const float* w0b    = F(18);
  const float* a0     = F(19);
  const float* a1     = F(20);
  const float* a2     = F(21);
  const float* v0     = F(22);
  const float* v1     = F(23);
  const float* v2     = F(24);
  const float* g1     = F(25);
  const float* g2     = F(26);
  const float* k_k    = F(27);
  const float* k_a    = F(28);
  const float* r_k    = F(29);
  const float* kw     = F(30);
  const float* vw     = F(31);
  const float* rw     = F(32);
  const float* ow     = F(33);
  const float* gnw    = F(34);
  const float* gnb    = F(35);
  const float* ffnxk  = F(36);
  const float* ffnkw  = F(37);
  const float* ffnvw  = F(38);
  const float* sxatt  = F(39);
  const float* sstate = F(40);
  const float* sxffn  = F(41);

  // Output layout: logits[V] | na[L*C] | nS[L*H*N*N] | nf[L*C]
  float* out_logits = (float*)d_out;
  float* out_na = out_logits + kV;
  float* out_nS = out_na + kL * kC;
  float* out_nf = out_nS + kL * kH * kN * kN;

  // Workspace layout (1024-float slots)
  float* WS = (float*)d_ws;
  float* X     = WS + 0 * kC;
  float* XR    = WS + 1 * kC;
  float* XW    = WS + 2 * kC;
  float* XK    = WS + 3 * kC;
  float* XV    = WS + 4 * kC;
  float* XA    = WS + 5 * kC;
  float* XG    = WS + 6 * kC;
  float* R     = WS + 7 * kC;
  float* K_    = WS + 8 * kC;
  float* Vv    = WS + 9 * kC;
  float* Wd    = WS + 10 * kC;
  float* A     = WS + 11 * kC;
  float* G     = WS + 12 * kC;
  float* VG    = WS + 13 * kC;
  float* VF    = WS + 14 * kC;
  float* KK    = WS + 15 * kC;
  float* ATT   = WS + 16 * kC;
  float* ATT2  = WS + 17 * kC;
  float* FFIN  = WS + 18 * kC;
  float* FFOUT = WS + 19 * kC;
  float* TW    = WS + 20 * kC;        // 64
  float* TA    = WS + 20 * kC + 64;   // 64
  float* TV    = WS + 20 * kC + 128;  // 32
  float* TG    = WS + 20 * kC + 160;  // 128
  float* FFK   = WS + 21 * kC;        // 4096
  float* XFIN  = WS + 25 * kC;

  embed_kernel<<<4, 256, 0, stream>>>(emb, token, X);

  const float* resid = nullptr;
  for (int i = 0; i < kL; ++i) {
    ln_mix6_kernel<<<1, 256, 0, stream>>>(
        X, resid, ln1w + i * kC, ln1b + i * kC, sxatt + i * kC,
        xrm + i * kC, xwm + i * kC, xkm + i * kC,
        xvm + i * kC, xam + i * kC, xgm + i * kC,
        XR, XW, XK, XV, XA, XG, out_na + i * kC);

    // big GEMVs (bandwidth-bound): r, k, v
    gemv_kernel<<<kC / 8, 256, 0, stream>>>(rw + (size_t)i * kC * kC, XR, R,  nullptr, kC, kC, 0);
    gemv_kernel<<<kC / 8, 256, 0, stream>>>(kw + (size_t)i * kC * kC, XK, K_, nullptr, kC, kC, 0);
    gemv_kernel<<<kC / 8, 256, 0, stream>>>(vw + (size_t)i * kC * kC, XV, Vv, nullptr, kC, kC, 0);

    // lora chains
    gemv_kernel<<<kLW / 8, 256, 0, stream>>>(w0 + (size_t)i * kLW * kC, XW, TW, nullptr, kLW, kC, 1); // tanh
    gemv_kernel<<<kC / 8, 256, 0, stream>>>(w1 + (size_t)i * kC * kLW, TW, Wd, w2 + i * kC, kC, kLW, 0);
    gemv_kernel<<<kLA / 8, 256, 0, stream>>>(a0 + (size_t)i * kLA * kC, XA, TA, nullptr, kLA, kC, 0);
    gemv_kernel<<<kC / 8, 256, 0, stream>>>(a1 + (size_t)i * kC * kLA, TA, A, a2 + i * kC, kC, kLA, 2); // sigmoid
    if (i > 0) {
      gemv_kernel<<<kLV / 8, 256, 0, stream>>>(v0 + (size_t)i * kLV * kC, XV, TV, nullptr, kLV, kC, 0);
      gemv_kernel<<<kC / 8, 256, 0, stream>>>(v1 + (size_t)i * kC * kLV, TV, VG, v2 + i * kC, kC, kLV, 2);
    }
    gemv_kernel<<<kLG / 8, 256, 0, stream>>>(g1 + (size_t)i * kLG * kC, XG, TG, nullptr, kLG, kC, 2); // sigmoid
    gemv_kernel<<<kC / 8, 256, 0, stream>>>(g2 + (size_t)i * kC * kLG, TG, G, nullptr, kC, kLG, 0);

    prep_state_kernel<<<1, 256, 0, stream>>>(
        K_, k_k + i * kC, A, k_a + i * kC, Vv, VG, VF, Wd, w0b + i * kC, KK,
        (i == 0) ? 1 : 0);

    state_update_kernel<<<kH, 32, 0, stream>>>(
        sstate + (size_t)i * kH * kN * kN, out_nS + (size_t)i * kH * kN * kN,
        R, K_, Vv, KK, A, Wd, ATT, gnw + i * kC, gnb + i * kC, r_k + i * kC, G);

    gemv_kernel<<<kC / 8, 256, 0, stream>>>(ow + (size_t)i * kC * kC, ATT, ATT2, nullptr, kC, kC, 0);

    ln_mix1_kernel<<<1, 256, 0, stream>>>(
        X, ATT2, ln2w + i * kC, ln2b + i * kC, sxffn + i * kC, ffnxk + i * kC,
        out_nf + i * kC, FFIN);

    gemv_kernel<<<kFF / 8, 256, 0, stream>>>(ffnkw + (size_t)i * kFF * kC, FFIN, FFK, nullptr, kFF, kC, 3); // relu^2
    gemv_kernel<<<kC / 8, 256, 0, stream>>>(ffnvw + (size_t)i * kC * kFF, FFK, FFOUT, nullptr, kC, kFF, 0);

    resid = FFOUT;
  }

  ln_final_kernel<<<1, 256, 0, stream>>>(X, FFOUT, lnow, lnob, XFIN);
  gemv_kernel<<<kV / 8, 256, 0, stream>>>(head_w, XFIN, out_logits, nullptr, kV, kC, 0);
}